// MotionGRU_56521769615775
// MI455X (gfx1250) — compile-verified
//
#include <hip/hip_runtime.h>
#include <math.h>

#define NB 4
#define NN 4096
#define KK 16
#define HIDC 128

typedef float v2f __attribute__((ext_vector_type(2)));
typedef float v8f __attribute__((ext_vector_type(8)));

// ---------------------------------------------------------------------------
// Kernel 1: brute-force KNN (top-16) + relative coords (nn_group)
// ---------------------------------------------------------------------------
__global__ __launch_bounds__(256) void knn_kernel(
    const float* __restrict__ points0,   // [B,3,N]
    const float* __restrict__ points1,   // [B,3,N]
    int*   __restrict__ nn_idx,          // [B,N,16]
    float* __restrict__ nn_group)        // [B,N,16,4]
{
    __shared__ float px[NN], py[NN], pz[NN];
    const int b = blockIdx.y;
    const int t = threadIdx.x;
    const float* p0 = points0 + (long long)b * 3 * NN;
    for (int i = t; i < NN; i += 256) {
        px[i] = p0[i];
        py[i] = p0[NN + i];
        pz[i] = p0[2 * NN + i];
    }
    __syncthreads();

    const int n  = blockIdx.x * 256 + t;
    const float ax = points1[(long long)b * 3 * NN + n];
    const float ay = points1[(long long)b * 3 * NN + NN + n];
    const float az = points1[(long long)b * 3 * NN + 2 * NN + n];

    float bd[KK]; int bi[KK];
#pragma unroll
    for (int j = 0; j < KK; ++j) { bd[j] = 3.4e38f; bi[j] = 0; }

    for (int i = 0; i < NN; ++i) {
        float dx = ax - px[i], dy = ay - py[i], dz = az - pz[i];
        float d = dx * dx + dy * dy + dz * dz;
        if (d < bd[0]) {                 // bd[0] is current worst of the 16
            bd[0] = d; bi[0] = i;
            // restore "slot 0 = max" invariant with static-index swap chain
#pragma unroll
            for (int j = 1; j < KK; ++j) {
                if (bd[j] > bd[0]) {
                    float td = bd[0]; bd[0] = bd[j]; bd[j] = td;
                    int   ti = bi[0]; bi[0] = bi[j]; bi[j] = ti;
                }
            }
        }
    }

    const long long base = ((long long)b * NN + n) * KK;
#pragma unroll
    for (int j = 0; j < KK; ++j) {
        int id = bi[j];
        nn_idx[base + j] = id;
        float rx = px[id] - ax, ry = py[id] - ay, rz = pz[id] - az;
        float dist = sqrtf(rx * rx + ry * ry + rz * rz);
        float* g = nn_group + (base + j) * 4;
        g[0] = rx; g[1] = ry; g[2] = rz; g[3] = dist;
    }
}

// ---------------------------------------------------------------------------
// Kernel 2: stacked [W_R|W_Z|W_H0][:, :132] (384x132) @ x_h(132 x N*K),
//           max over the K=16 neighbors, via V_WMMA_F32_16X16X4_F32.
//           One B-tile = the 16 neighbors of one anchor -> max = lane butterfly.
// ---------------------------------------------------------------------------
__global__ __launch_bounds__(256) void gemm_knn_kernel(
    const float* __restrict__ H0,        // [B,128,N]
    const float* __restrict__ W_R,       // [128,260]
    const float* __restrict__ W_Z,       // [128,260]
    const float* __restrict__ W_H0,      // [128,132]
    const int*   __restrict__ nn_idx,    // [B,N,16]
    const float* __restrict__ nn_group,  // [B,N,16,4]
    float* __restrict__ pre)             // [B,384,N]
{
    __shared__ float sW[64 * 132];       // 64-channel weight slice
    const int b = blockIdx.z;
    const int g = blockIdx.y;            // channel group 0..5 (64 ch each)
    const int chunk = blockIdx.x;        // 64 anchors per block
    const int t = threadIdx.x;

    for (int i = t; i < 64 * 132; i += 256) {
        int cp = i / 132, j = i % 132;
        int row = g * 64 + cp;
        float w;
        if (row < 128)      w = W_R[row * 260 + j];
        else if (row < 256) w = W_Z[(row - 128) * 260 + j];
        else                w = W_H0[(row - 256) * 132 + j];
        sW[i] = w;
    }
    __syncthreads();

    const int wave = t >> 5;
    const int lane = t & 31;
    const int col  = lane & 15;          // neighbor index k (B cols) / ch row (A)
    const int hi   = lane >> 4;          // 0: K={0,1}, 1: K={2,3} per WMMA layout

    const float* h0b = H0 + (long long)b * HIDC * NN;

    for (int it = 0; it < 8; ++it) {
        const int n = chunk * 64 + it * 8 + wave;   // this wave's anchor
        const long long nb = ((long long)b * NN + n) * KK + col;
        const int idx = nn_idx[nb];

        // B regs: x_h[132 x 16] striped per ISA f32 B layout (VGPR0=K0/K2, VGPR1=K1/K3)
        v2f Breg[33];
        {
            const float* gs = nn_group + nb * 4 + hi * 2;
            v2f v; v[0] = gs[0]; v[1] = gs[1];
            Breg[0] = v;
        }
#pragma unroll
        for (int s = 1; s < 33; ++s) {
            int c0 = 4 * s - 4 + hi * 2;
            v2f v; v[0] = h0b[(long long)c0 * NN + idx];
            v[1] = h0b[(long long)(c0 + 1) * NN + idx];
            Breg[s] = v;
        }

        for (int m = 0; m < 4; ++m) {
            v8f acc = {};
            const float* wb = sW + (m * 16 + col) * 132 + hi * 2;
#pragma unroll
            for (int s = 0; s < 33; ++s) {
                v2f a; a[0] = wb[4 * s]; a[1] = wb[4 * s + 1];
                acc = __builtin_amdgcn_wmma_f32_16x16x4_f32(
                    false, a, false, Breg[s], (short)0, acc, false, false);
            }
            // max over the 16 neighbor columns (lanes within each 16-lane half)
#pragma unroll
            for (int e = 0; e < 8; ++e) {
                float v = acc[e];
                v = fmaxf(v, __shfl_xor(v, 1, 32));
                v = fmaxf(v, __shfl_xor(v, 2, 32));
                v = fmaxf(v, __shfl_xor(v, 4, 32));
                v = fmaxf(v, __shfl_xor(v, 8, 32));
                acc[e] = v;
            }
            if (col == 0) {              // lanes 0 (rows 0-7) and 16 (rows 8-15)
#pragma unroll
                for (int e = 0; e < 8; ++e) {
                    int c = g * 64 + m * 16 + e + hi * 8;
                    pre[((long long)b * 384 + c) * NN + n] = acc[e];
                }
            }
        }
    }
}

// ---------------------------------------------------------------------------
// Kernel 3: [W_Rf; W_Zf] (256x128) @ feats (128xN) per batch (feats = broadcast
//           part pulled out of the max) -> frz [B,256,N]
// ---------------------------------------------------------------------------
__global__ __launch_bounds__(256) void gemm_feats_kernel(
    const float* __restrict__ contents1, // [B,64,N]
    const float* __restrict__ motions1,  // [B,64,N]
    const float* __restrict__ W_R,
    const float* __restrict__ W_Z,
    float* __restrict__ frz)             // [B,256,N]
{
    __shared__ float sW[64 * 128];
    const int b = blockIdx.z;
    const int g = blockIdx.y;            // 0..3 (64 channels each)
    const int chunk = blockIdx.x;        // 128 anchors per block
    const int t = threadIdx.x;

    for (int i = t; i < 64 * 128; i += 256) {
        int cp = i / 128, j = i % 128;
        int row = g * 64 + cp;
        sW[i] = (row < 128) ? W_R[row * 260 + 132 + j]
                            : W_Z[(row - 128) * 260 + 132 + j];
    }
    __syncthreads();

    const int wave = t >> 5, lane = t & 31, col = lane & 15, hi = lane >> 4;
    const int n = chunk * 128 + wave * 16 + col;
    const float* c1 = contents1 + (long long)b * 64 * NN;
    const float* m1 = motions1  + (long long)b * 64 * NN;

    v2f Breg[32];
#pragma unroll
    for (int s = 0; s < 32; ++s) {
        int c0 = 4 * s + hi * 2;                      // even -> no boundary straddle
        const float* src = (c0 < 64) ? (c1 + (long long)c0 * NN)
                                     : (m1 + (long long)(c0 - 64) * NN);
        v2f v; v[0] = src[n]; v[1] = src[NN + n];
        Breg[s] = v;
    }

    for (int m = 0; m < 4; ++m) {
        v8f acc = {};
        const float* wb = sW + (m * 16 + col) * 128 + hi * 2;
#pragma unroll
        for (int s = 0; s < 32; ++s) {
            v2f a; a[0] = wb[4 * s]; a[1] = wb[4 * s + 1];
            acc = __builtin_amdgcn_wmma_f32_16x16x4_f32(
                false, a, false, Breg[s], (short)0, acc, false, false);
        }
#pragma unroll
        for (int e = 0; e < 8; ++e) {
            int c = g * 64 + m * 16 + e + hi * 8;
            frz[((long long)b * 256 + c) * NN + n] = acc[e];
        }
    }
}

// ---------------------------------------------------------------------------
// Kernel 4: gates (sigmoid) + H1_0 bias + u = gate_R * H1_0
// ---------------------------------------------------------------------------
__global__ __launch_bounds__(256) void gate_kernel(
    const float* __restrict__ pre, const float* __restrict__ frz,
    const float* __restrict__ b_R, const float* __restrict__ b_Z,
    const float* __restrict__ b_H0,
    float* __restrict__ u, float* __restrict__ gz, float* __restrict__ h10)
{
    long long i = (long long)blockIdx.x * 256 + threadIdx.x;  // over B*128*N
    long long n = i % NN;
    long long c = (i / NN) % HIDC;
    long long b = i / ((long long)HIDC * NN);
    float pR = pre[(b * 384 +       c) * NN + n];
    float pZ = pre[(b * 384 + 128 + c) * NN + n];
    float pH = pre[(b * 384 + 256 + c) * NN + n];
    float fR = frz[(b * 256 +       c) * NN + n];
    float fZ = frz[(b * 256 + 128 + c) * NN + n];
    float gR  = 1.0f / (1.0f + expf(-(pR + fR + b_R[c])));
    float gZv = 1.0f / (1.0f + expf(-(pZ + fZ + b_Z[c])));
    float h   = pH + b_H0[c];
    u[i]   = gR * h;
    gz[i]  = gZv;
    h10[i] = h;
}

// ---------------------------------------------------------------------------
// Kernel 5: H1_1 = tanh(W_H1 @ [feats; u] + b_H1); out = gz*h10 + (1-gz)*H1_1
// ---------------------------------------------------------------------------
__global__ __launch_bounds__(256) void gemm_out_kernel(
    const float* __restrict__ contents1, const float* __restrict__ motions1,
    const float* __restrict__ u, const float* __restrict__ gz,
    const float* __restrict__ h10,
    const float* __restrict__ W_H1, const float* __restrict__ b_H1,
    float* __restrict__ out)             // [B,128,N]
{
    __shared__ float sW[32 * 256];
    const int b = blockIdx.z;
    const int g = blockIdx.y;            // 0..3 (32 channels each)
    const int chunk = blockIdx.x;        // 128 anchors per block
    const int t = threadIdx.x;

    for (int i = t; i < 32 * 256; i += 256) {
        int cp = i / 256, j = i % 256;
        sW[i] = W_H1[(g * 32 + cp) * 256 + j];
    }
    __syncthreads();

    const int wave = t >> 5, lane = t & 31, col = lane & 15, hi = lane >> 4;
    const int n = chunk * 128 + wave * 16 + col;
    const float* c1 = contents1 + (long long)b * 64 * NN;
    const float* m1 = motions1  + (long long)b * 64 * NN;
    const float* ub = u + (long long)b * HIDC * NN;

    v2f Breg[64];
#pragma unroll
    for (int s = 0; s < 64; ++s) {
        int c0 = 4 * s + hi * 2;
        const float* src;
        if (c0 < 64)       src = c1 + (long long)c0 * NN;
        else if (c0 < 128) src = m1 + (long long)(c0 - 64) * NN;
        else               src = ub + (long long)(c0 - 128) * NN;
        v2f v; v[0] = src[n]; v[1] = src[NN + n];
        Breg[s] = v;
    }

    for (int m = 0; m < 2; ++m) {
        v8f acc = {};
        const float* wb = sW + (m * 16 + col) * 256 + hi * 2;
#pragma unroll
        for (int s = 0; s < 64; ++s) {
            v2f a; a[0] = wb[4 * s]; a[1] = wb[4 * s + 1];
            acc = __builtin_amdgcn_wmma_f32_16x16x4_f32(
                false, a, false, Breg[s], (short)0, acc, false, false);
        }
#pragma unroll
        for (int e = 0; e < 8; ++e) {
            int c = g * 32 + m * 16 + e + hi * 8;
            long long o = ((long long)b * HIDC + c) * NN + n;
            float h11 = tanhf(acc[e] + b_H1[c]);
            float z = gz[o];
            out[o] = z * h10[o] + (1.0f - z) * h11;
        }
    }
}

// ---------------------------------------------------------------------------
// Launch. Workspace layout (floats unless noted), total ~72.4 MB:
//   [0, 262144)            nn_idx (int)
//   [262144, 1310720)      nn_group
//   [1310720, 7602176)     pre   [B,384,N]
//   [7602176, 11796480)    frz   [B,256,N]
//   [11796480, 13893632)   u     [B,128,N]
//   [13893632, 15990784)   gz    [B,128,N]
//   [15990784, 18087936)   h10   [B,128,N]
// ---------------------------------------------------------------------------
extern "C" void kernel_launch(void* const* d_in, const int* in_sizes, int n_in,
                              void* d_out, int out_size, void* d_ws, size_t ws_size,
                              hipStream_t stream) {
    (void)in_sizes; (void)n_in; (void)out_size; (void)ws_size;
    const float* H0        = (const float*)d_in[0];
    const float* points0   = (const float*)d_in[1];
    const float* points1   = (const float*)d_in[2];
    const float* contents1 = (const float*)d_in[3];
    const float* motions1  = (const float*)d_in[4];
    const float* W_R  = (const float*)d_in[5];
    const float* b_R  = (const float*)d_in[6];
    const float* W_Z  = (const float*)d_in[7];
    const float* b_Z  = (const float*)d_in[8];
    const float* W_H0 = (const float*)d_in[9];
    const float* b_H0 = (const float*)d_in[10];
    const float* W_H1 = (const float*)d_in[11];
    const float* b_H1 = (const float*)d_in[12];
    float* out = (float*)d_out;

    float* wsf      = (float*)d_ws;
    int*   nn_idx   = (int*)wsf;
    float* nn_group = wsf + 262144;
    float* pre      = wsf + 1310720;
    float* frz      = wsf + 7602176;
    float* u        = wsf + 11796480;
    float* gz       = wsf + 13893632;
    float* h10      = wsf + 15990784;

    knn_kernel<<<dim3(16, 4), 256, 0, stream>>>(points0, points1, nn_idx, nn_group);
    gemm_knn_kernel<<<dim3(64, 6, 4), 256, 0, stream>>>(H0, W_R, W_Z, W_H0,
                                                        nn_idx, nn_group, pre);
    gemm_feats_kernel<<<dim3(32, 4, 4), 256, 0, stream>>>(contents1, motions1,
                                                          W_R, W_Z, frz);
    gate_kernel<<<dim3(8192), 256, 0, stream>>>(pre, frz, b_R, b_Z, b_H0, u, gz, h10);
    gemm_out_kernel<<<dim3(32, 4, 4), 256, 0, stream>>>(contents1, motions1, u, gz,
                                                        h10, W_H1, b_H1, out);
}